// QuantumLayer_85074712199264
// MI455X (gfx1250) — compile-verified
//
#include <hip/hip_runtime.h>
#include <math.h>

#define NQ   12
#define DIM  4096          // 2^12
#define BLK  256           // 8 wave32s

typedef __attribute__((ext_vector_type(2))) float v2f;
typedef __attribute__((ext_vector_type(8))) float v8f;

#if defined(__has_builtin)
#if __has_builtin(__builtin_amdgcn_wmma_f32_16x16x4_f32)
#define HAVE_WMMA_F32 1
#endif
#endif

struct cpx { float r, i; };
__device__ __forceinline__ cpx cmul(cpx a, cpx b) {
    return { a.r * b.r - a.i * b.i, a.r * b.i + a.i * b.r };
}
__device__ __forceinline__ cpx cadd(cpx a, cpx b) { return { a.r + b.r, a.i + b.i }; }

__global__ __launch_bounds__(BLK)
void qsim_kernel(const float* __restrict__ x, const float* __restrict__ w,
                 float* __restrict__ out)
{
    __shared__ float2 s[DIM];          // state vector (re, im), 32 KB
    __shared__ float  gates[24][8];    // fused RZ*RY*RX per (layer, wire)
    __shared__ float  trigc[NQ], trigs[NQ];
    __shared__ float  T[64][16];       // WMMA partial projection

    const int t = threadIdx.x;
    const int b = blockIdx.x;

    // ---- setup: encoding trig + fused layer gates ----
    if (t < NQ) {
        float sv, cv;
        __sincosf(0.5f * x[b * NQ + t], &sv, &cv);
        trigc[t] = cv; trigs[t] = sv;
    }
    if (t < 24) {
        const int layer = t / NQ, wi = t % NQ;
        const int base = 3 * (layer * NQ + wi);
        float sx, cx, sy, cy, sz, cz;
        __sincosf(0.5f * w[base + 0], &sx, &cx);
        __sincosf(0.5f * w[base + 1], &sy, &cy);
        __sincosf(0.5f * w[base + 2], &sz, &cz);
        // RX = [[(cx,0),(0,-sx)],[(0,-sx),(cx,0)]], RY = [[(cy,0),(-sy,0)],[(sy,0),(cy,0)]]
        cpx RX[2][2] = { { {cx,0}, {0,-sx} }, { {0,-sx}, {cx,0} } };
        cpx RY[2][2] = { { {cy,0}, {-sy,0} }, { {sy,0}, {cy,0} } };
        cpx G1[2][2];
        #pragma unroll
        for (int r = 0; r < 2; ++r)
            #pragma unroll
            for (int c = 0; c < 2; ++c)
                G1[r][c] = cadd(cmul(RY[r][0], RX[0][c]), cmul(RY[r][1], RX[1][c]));
        // RZ = diag(cz - i*sz, cz + i*sz) row-scales G1
        cpx e0 = { cz, -sz }, e1 = { cz, sz };
        cpx G[2][2] = { { cmul(e0, G1[0][0]), cmul(e0, G1[0][1]) },
                        { cmul(e1, G1[1][0]), cmul(e1, G1[1][1]) } };
        float* g = gates[t];
        g[0] = G[0][0].r; g[1] = G[0][0].i; g[2] = G[0][1].r; g[3] = G[0][1].i;
        g[4] = G[1][0].r; g[5] = G[1][0].i; g[6] = G[1][1].r; g[7] = G[1][1].i;
    }
    __syncthreads();

    // ---- initial product state from RY(x) applied to |0..0> (real) ----
    #pragma unroll
    for (int q = 0; q < 16; ++q) {
        const int i = t + (q << 8);
        float prod = 1.f;
        #pragma unroll
        for (int wi = 0; wi < NQ; ++wi)
            prod *= ((i >> (11 - wi)) & 1) ? trigs[wi] : trigc[wi];
        s[i] = make_float2(prod, 0.f);
    }
    __syncthreads();

    // ---- 2 layers: 12 fused 1q gates + CNOT-chain permutation ----
    for (int layer = 0; layer < 2; ++layer) {
        for (int wi = 0; wi < NQ; ++wi) {
            const float* g = gates[layer * NQ + wi];
            const float g00r = g[0], g00i = g[1], g01r = g[2], g01i = g[3];
            const float g10r = g[4], g10i = g[5], g11r = g[6], g11i = g[7];
            const int bp = 11 - wi;
            #pragma unroll
            for (int q = 0; q < 8; ++q) {           // 2048 pairs / 256 threads
                const int p  = t + (q << 8);
                const int i0 = ((p >> bp) << (bp + 1)) | (p & ((1 << bp) - 1));
                const int i1 = i0 | (1 << bp);
                const float2 a0 = s[i0], a1 = s[i1];
                const float nr0 = g00r*a0.x - g00i*a0.y + g01r*a1.x - g01i*a1.y;
                const float ni0 = g00r*a0.y + g00i*a0.x + g01r*a1.y + g01i*a1.x;
                const float nr1 = g10r*a0.x - g10i*a0.y + g11r*a1.x - g11i*a1.y;
                const float ni1 = g10r*a0.y + g10i*a0.x + g11r*a1.y + g11i*a1.x;
                s[i0] = make_float2(nr0, ni0);
                s[i1] = make_float2(nr1, ni1);
            }
            __syncthreads();
        }
        // CNOT chain w=0..10 collapses to: s_new[j] = s_old[j ^ (j>>1)]
        float2 tmp[16];
        #pragma unroll
        for (int q = 0; q < 16; ++q) {
            const int j = t + (q << 8);
            tmp[q] = s[j ^ (j >> 1)];
        }
        __syncthreads();
        #pragma unroll
        for (int q = 0; q < 16; ++q) s[t + (q << 8)] = tmp[q];
        __syncthreads();
    }

    // ---- probabilities in place (into .x) ----
    #pragma unroll
    for (int q = 0; q < 16; ++q) {
        const int i = t + (q << 8);
        const float2 a = s[i];
        s[i].x = a.x * a.x + a.y * a.y;
    }
    __syncthreads();

#ifdef HAVE_WMMA_F32
    // ---- projection: T[hi][n] = sum_lo p[hi*64+lo] * S[lo][n] via WMMA f32 16x16x4.
    // Cols 0..5: signs for low wires 6..11 (bit 5-n of lo); col 6: ones (row sums
    // feeding the high wires); cols 7..15: zero.
    const int wv   = t >> 5;           // wave id (0..7)
    const int lane = t & 31;
    if (wv < 4) {                      // wave-uniform branch; EXEC all ones inside
        const int n     = lane & 15;
        const int half  = lane >> 4;
        const int kOff  = half * 2;    // lanes 0-15: K=0,1 ; lanes 16-31: K=2,3
        const int myRow = wv * 16 + n; // A-matrix M index
        v8f acc = { 0.f, 0.f, 0.f, 0.f, 0.f, 0.f, 0.f, 0.f };
        #pragma unroll
        for (int step = 0; step < 16; ++step) {
            const int kb = step * 4;
            v2f A, B;
            A.x = s[myRow * 64 + kb + kOff].x;
            A.y = s[myRow * 64 + kb + kOff + 1].x;
            #pragma unroll
            for (int j = 0; j < 2; ++j) {
                const int lo = kb + kOff + j;
                float v;
                if (n < 6)       v = ((lo >> (5 - n)) & 1) ? -1.f : 1.f;
                else if (n == 6) v = 1.f;
                else             v = 0.f;
                if (j == 0) B.x = v; else B.y = v;
            }
            acc = __builtin_amdgcn_wmma_f32_16x16x4_f32(
                      false, A, false, B, (short)0, acc, false, false);
        }
        #pragma unroll
        for (int j = 0; j < 8; ++j)
            T[wv * 16 + j + half * 8][n] = acc[j];
    }
    __syncthreads();

    if (t < NQ) {
        float acc = 0.f;
        if (t >= 6) {                       // low wires: column sums of T
            const int col = t - 6;
            for (int hi = 0; hi < 64; ++hi) acc += T[hi][col];
        } else {                            // high wires: signed sum of row-sums
            const int bp = 5 - t;
            for (int hi = 0; hi < 64; ++hi) {
                const float v = T[hi][6];
                acc += ((hi >> bp) & 1) ? -v : v;
            }
        }
        out[b * NQ + t] = acc;
    }
#else
    // VALU fallback: direct signed reduction
    if (t < NQ) {
        float acc = 0.f;
        const int bp = 11 - t;
        for (int i = 0; i < DIM; ++i) {
            const float p = s[i].x;
            acc += ((i >> bp) & 1) ? -p : p;
        }
        out[b * NQ + t] = acc;
    }
#endif
}

extern "C" void kernel_launch(void* const* d_in, const int* in_sizes, int n_in,
                              void* d_out, int out_size, void* d_ws, size_t ws_size,
                              hipStream_t stream) {
    (void)n_in; (void)out_size; (void)d_ws; (void)ws_size;
    const float* x   = (const float*)d_in[0];
    const float* wts = (const float*)d_in[1];
    float*       out = (float*)d_out;
    const int batch = in_sizes[0] / NQ;     // 8192
    qsim_kernel<<<batch, BLK, 0, stream>>>(x, wts, out);
}